// Actor_59708635349236
// MI455X (gfx1250) — compile-verified
//
#include <hip/hip_runtime.h>
#include <hip/hip_bf16.h>
#include <stdint.h>

typedef __attribute__((ext_vector_type(2))) float v2f;
typedef __attribute__((ext_vector_type(4))) float v4f;
typedef __attribute__((ext_vector_type(8))) float v8f;

// builtin-matching pointer types for the async global->LDS ops
typedef int vsi4 __attribute__((__vector_size__(16)));                 // V4i
typedef __attribute__((address_space(1))) vsi4* g128_t;                // global src, b128
typedef __attribute__((address_space(3))) vsi4* l128_t;                // LDS dst,   b128
typedef __attribute__((address_space(1))) int*  g32_t;                 // global src, b32
typedef __attribute__((address_space(3))) int*  l32_t;                 // LDS dst,   b32

// CDNA5 async global->LDS path (ASYNCcnt), guarded so absence of the builtins
// falls back to the register-staged path instead of breaking the build.
#if defined(__has_builtin)
#  if __has_builtin(__builtin_amdgcn_global_load_async_to_lds_b128) && \
      __has_builtin(__builtin_amdgcn_global_load_async_to_lds_b32)  && \
      __has_builtin(__builtin_amdgcn_s_wait_asynccnt)
#    define ASYNC_LDS 1
#  endif
#endif
#ifndef ASYNC_LDS
#  define ASYNC_LDS 0
#endif

#define B_DIM 4096
#define N_DIM 16384
#define D_DIM 256
#define H_DIM 1024

#define TM 128
#define TN 128
#define TK 32
#define ASTRIDE 36          // A tile row stride (floats): 144B -> 16B-aligned b128, banks spread
#define NKP (TK / 2)        // 16 K-pair rows in B tile
#define BPST 288            // B pair-row stride (floats): == 32 mod 64 -> conflict-free b64 reads

// ---------------------------------------------------------------------------
// fp32 WMMA GEMM: D = act(A@B [+ A2@B2] + bias), optional mask -> -inf
// WG: 256 threads = 8 waves (4 along M x 2 along N). Wave tile 32x64.
// LDS B tile is K-paired: Bs[k>>1][n][k&1] so a lane's B operand
// ( B[k][n], B[k+1][n] ) is one aligned ds_load_b64 -> no repack moves.
// Async addressing: per-lane offsets are loop-invariant, tile-varying parts
// are uniform -> backend can use saddr + 32-bit voffset form, no per-load muls.
// ---------------------------------------------------------------------------
template<bool DUAL, bool RELU, bool MASKED>
__global__ __launch_bounds__(256) void wmma_gemm_f32(
    const float* __restrict__ A,  const float* __restrict__ B,
    const float* __restrict__ bias,
    const float* __restrict__ A2, const float* __restrict__ B2,
    const uint8_t* __restrict__ mask,
    float* __restrict__ D,
    int M, int N, int K, int K2)
{
  __shared__ float As[2][TM * ASTRIDE];
  __shared__ float Bs[2][NKP * BPST];

  const int tid   = threadIdx.x;
  const int lane  = tid & 31;
  const int wave  = tid >> 5;
  const int wm    = wave & 3;        // wave row  (4 along M)
  const int wn    = wave >> 2;       // wave col  (2 along N)
  const int m0    = blockIdx.y * TM;
  const int n0    = blockIdx.x * TN;
  const int lhalf = lane >> 4;       // 0: lanes 0-15, 1: lanes 16-31
  const int l15   = lane & 15;

  v8f acc[2][4];
  {
    v8f z = {};
    #pragma unroll
    for (int mi = 0; mi < 2; ++mi)
      #pragma unroll
      for (int ni = 0; ni < 4; ++ni) acc[mi][ni] = z;
  }

  const int t1 = K / TK;
  const int T  = t1 + (DUAL ? (K2 / TK) : 0);

  auto compute_tile = [&](const float* __restrict__ Asb, const float* __restrict__ Bsb) {
    #pragma unroll
    for (int kk = 0; kk < TK; kk += 4) {
      v2f af[2], bf[4];
      #pragma unroll
      for (int mi = 0; mi < 2; ++mi) {
        // A 16x4 f32 layout: lanes 0-15 -> K=kk,kk+1 ; lanes 16-31 -> K=kk+2,kk+3
        const int row = wm * 32 + mi * 16 + l15;
        af[mi] = *(const v2f*)(Asb + row * ASTRIDE + kk + lhalf * 2);
      }
      #pragma unroll
      for (int ni = 0; ni < 4; ++ni) {
        // one b64: { B[kr][col], B[kr+1][col] }, kr = kk + 2*lhalf
        const int col = wn * 64 + ni * 16 + l15;
        bf[ni] = *(const v2f*)(Bsb + ((kk >> 1) + lhalf) * BPST + col * 2);
      }
      #pragma unroll
      for (int mi = 0; mi < 2; ++mi)
        #pragma unroll
        for (int ni = 0; ni < 4; ++ni)
          acc[mi][ni] = __builtin_amdgcn_wmma_f32_16x16x4_f32(
              false, af[mi], false, bf[ni], (short)0, acc[mi][ni], false, false);
    }
  };

#if ASYNC_LDS
  // ---- double-buffered async global->LDS pipeline (ASYNCcnt) ----
  // per-lane invariants (computed once)
  const int arow  = tid >> 3;                 // A tile row base within a j-slice
  const int acol  = (tid & 7) * 4;            // A tile col (float4 granularity)
  const int alnA  = arow * K + acol;          // per-lane A offset, pass 1
  const int alnB  = DUAL ? (arow * K2 + acol) : 0;   // per-lane A offset, pass 2
  const int bn    = tid & 127;                // B tile col
  const int bk    = tid >> 7;                 // B tile k parity base (0..1)
  const int bln   = bk * N + n0 + bn;         // per-lane B offset (loop-invariant)
  const int blds  = bn * 2 + bk;              // per-lane B LDS index part

  auto issue_tile = [&](int t, int buf) {
    const float* Ap; const float* Bp; int Kp; int k0; int alane;
    if (!DUAL || t < t1) { Ap = A;  Bp = B;  Kp = K;  k0 = t * TK;        alane = alnA; }
    else                 { Ap = A2; Bp = B2; Kp = K2; k0 = (t - t1) * TK; alane = alnB; }
    #pragma unroll
    for (int j = 0; j < (TM * TK) / (256 * 4); ++j) {      // 4 x b128
      const float* gq = Ap + (m0 * Kp + k0 + (32 * j) * Kp);   // uniform (SGPR)
      float* lp = &As[buf][(arow + 32 * j) * ASTRIDE + acol];  // per-lane, invariant
      __builtin_amdgcn_global_load_async_to_lds_b128(
          (g128_t)(gq + alane), (l128_t)(lp), 0, 0);
    }
    #pragma unroll
    for (int j = 0; j < NKP; ++j) {                        // 16 x b32, K-paired scatter
      const float* gq = Bp + (k0 + 2 * j) * N;             // uniform (SGPR)
      float* lp = &Bs[buf][j * BPST + blds];               // per-lane, invariant
      __builtin_amdgcn_global_load_async_to_lds_b32(
          (g32_t)(gq + bln), (l32_t)(lp), 0, 0);
    }
  };

  issue_tile(0, 0);
  __builtin_amdgcn_s_wait_asynccnt(0);
  __syncthreads();
  for (int t = 0; t < T; ++t) {
    const int cb = t & 1;
    if (t + 1 < T) issue_tile(t + 1, cb ^ 1);   // fill other buffer while computing
    compute_tile(As[cb], Bs[cb]);
    if (t + 1 < T) {
      __builtin_amdgcn_s_wait_asynccnt(0);
      __syncthreads();
    }
  }
#else
  // ---- fallback: register-staged pipeline (single buffer) ----
  v4f avr[4], bvr[4];
  auto load_regs = [&](int t) {
    const float* Ap; const float* Bp; int Kp; int k0;
    if (!DUAL || t < t1) { Ap = A;  Bp = B;  Kp = K;  k0 = t * TK; }
    else                 { Ap = A2; Bp = B2; Kp = K2; k0 = (t - t1) * TK; }
    #pragma unroll
    for (int j = 0; j < 4; ++j) {
      const int fi = (tid + j * 256) * 4;
      const int r = fi >> 5, c = fi & 31;
      avr[j] = *(const v4f*)(Ap + (size_t)(m0 + r) * Kp + (k0 + c));
    }
    #pragma unroll
    for (int j = 0; j < 4; ++j) {
      const int fi = (tid + j * 256) * 4;
      const int r = fi >> 7, c = fi & 127;
      bvr[j] = *(const v4f*)(Bp + (size_t)(k0 + r) * N + (n0 + c));
    }
  };

  load_regs(0);
  for (int t = 0; t < T; ++t) {
    __syncthreads();
    #pragma unroll
    for (int j = 0; j < 4; ++j) {
      const int fi = (tid + j * 256) * 4;
      const int r = fi >> 5, c = fi & 31;
      *(v4f*)(&As[0][r * ASTRIDE + c]) = avr[j];
    }
    #pragma unroll
    for (int j = 0; j < 4; ++j) {
      const int fi = (tid + j * 256) * 4;
      const int r = fi >> 7, c = fi & 127;
      #pragma unroll
      for (int i = 0; i < 4; ++i)
        Bs[0][(r >> 1) * BPST + (c + i) * 2 + (r & 1)] = bvr[j][i];
    }
    __syncthreads();
    if (t + 1 < T) load_regs(t + 1);
    compute_tile(As[0], Bs[0]);
  }
#endif

  // epilogue: C layout -> vgpr i: lanes 0-15 = row (base+i), lanes 16-31 = row (base+8+i)
  const float NEG_INF = __int_as_float(0xff800000);
  #pragma unroll
  for (int mi = 0; mi < 2; ++mi) {
    #pragma unroll
    for (int ni = 0; ni < 4; ++ni) {
      const int n = n0 + wn * 64 + ni * 16 + l15;
      const float bvv = bias[n];
      #pragma unroll
      for (int i = 0; i < 8; ++i) {
        const int m = m0 + wm * 32 + mi * 16 + lhalf * 8 + i;
        float v = acc[mi][ni][i] + bvv;
        if (RELU) v = fmaxf(v, 0.0f);
        if (MASKED) { if (mask[(size_t)m * N + n]) v = NEG_INF; }
        D[(size_t)m * N + n] = v;
      }
    }
  }
}

// ---------------------------------------------------------------------------
// Per-row argmax of (logits + gumbel). One workgroup per row.
// First-occurrence tie-break to match jnp.argmax.
// ---------------------------------------------------------------------------
__global__ __launch_bounds__(256) void argmax_rows(
    const float* __restrict__ logits, const float* __restrict__ gumbel,
    int* __restrict__ idx, int N)
{
  __shared__ float sv[256];
  __shared__ int   si[256];
  const int row = blockIdx.x;
  const float* lr = logits + (size_t)row * N;
  const float* gr = gumbel + (size_t)row * N;
  float best = -__builtin_inff();
  int   bi   = 0;
  for (int n = threadIdx.x; n < N; n += 256) {
    float z = lr[n] + gr[n];          // masked entries are -inf and stay -inf
    if (z > best) { best = z; bi = n; }
  }
  sv[threadIdx.x] = best; si[threadIdx.x] = bi;
  __syncthreads();
  for (int s = 128; s > 0; s >>= 1) {
    if (threadIdx.x < s) {
      float v2 = sv[threadIdx.x + s]; int i2 = si[threadIdx.x + s];
      if (v2 > sv[threadIdx.x] || (v2 == sv[threadIdx.x] && i2 < si[threadIdx.x])) {
        sv[threadIdx.x] = v2; si[threadIdx.x] = i2;
      }
    }
    __syncthreads();
  }
  if (threadIdx.x == 0) idx[row] = si[0];
}

// ---------------------------------------------------------------------------
// Hard one-hot fill (straight-through forward value), float4 stores.
// grid: (N/1024, B)
// ---------------------------------------------------------------------------
__global__ __launch_bounds__(256) void onehot_fill(
    const int* __restrict__ idx, float* __restrict__ onehot, int N)
{
  const int row = blockIdx.y;
  const int c0  = (blockIdx.x * 256 + threadIdx.x) * 4;
  const int id  = idx[row];
  v4f v = {};
  if (id >= c0 && id < c0 + 4) v[id - c0] = 1.0f;
  *(v4f*)(onehot + (size_t)row * N + c0) = v;
}

// ---------------------------------------------------------------------------
// fragment[b,:] = frag_table[idx[b],:]   (onehot @ table == row gather)
// ---------------------------------------------------------------------------
__global__ __launch_bounds__(256) void gather_frag(
    const int* __restrict__ idx, const float* __restrict__ table,
    float* __restrict__ frag)
{
  const int gid = blockIdx.x * 256 + threadIdx.x;   // one float4 each
  const int b   = gid / (D_DIM / 4);
  const int off = (gid % (D_DIM / 4)) * 4;
  const int id  = idx[b];
  *(v4f*)(frag + (size_t)b * D_DIM + off) =
      *(const v4f*)(table + (size_t)id * D_DIM + off);
}

// ---------------------------------------------------------------------------
extern "C" void kernel_launch(void* const* d_in, const int* in_sizes, int n_in,
                              void* d_out, int out_size, void* d_ws, size_t ws_size,
                              hipStream_t stream)
{
  const float*   cond = (const float*)d_in[0];
  const uint8_t* mask = (const uint8_t*)d_in[1];   // jax bool -> 1 byte
  const float*   gum  = (const float*)d_in[2];
  const float*   ftab = (const float*)d_in[3];
  const float*   W1   = (const float*)d_in[4];
  const float*   b1   = (const float*)d_in[5];
  const float*   W2   = (const float*)d_in[6];
  const float*   b2   = (const float*)d_in[7];
  const float*   Wa   = (const float*)d_in[8];
  const float*   Wb   = (const float*)d_in[9];
  const float*   bm   = (const float*)d_in[10];

  float* logits = (float*)d_out;                                     // [B,N]
  float* onehot = logits + (size_t)B_DIM * N_DIM;                    // [B,N]
  float* frag   = onehot + (size_t)B_DIM * N_DIM;                    // [B,D]
  float* merger = frag   + (size_t)B_DIM * D_DIM;                    // [B,D]

  float* h   = (float*)d_ws;                                         // [B,H] 16 MB
  int*   idx = (int*)((char*)d_ws + (size_t)B_DIM * H_DIM * sizeof(float));

  // 1) h = relu(cond @ W1 + b1)
  wmma_gemm_f32<false, true, false><<<dim3(H_DIM / TN, B_DIM / TM), 256, 0, stream>>>(
      cond, W1, b1, nullptr, nullptr, nullptr, h, B_DIM, H_DIM, D_DIM, 0);

  // 2) logits = mask ? -inf : (h @ W2 + b2)
  wmma_gemm_f32<false, false, true><<<dim3(N_DIM / TN, B_DIM / TM), 256, 0, stream>>>(
      h, W2, b2, nullptr, nullptr, mask, logits, B_DIM, N_DIM, H_DIM, 0);

  // 3) idx = argmax_n(logits + gumbel)
  argmax_rows<<<B_DIM, 256, 0, stream>>>(logits, gum, idx, N_DIM);

  // 4) onehot = one_hot(idx)
  onehot_fill<<<dim3(N_DIM / 1024, B_DIM), 256, 0, stream>>>(idx, onehot, N_DIM);

  // 5) fragment = frag_table[idx]
  gather_frag<<<(B_DIM * D_DIM) / 1024, 256, 0, stream>>>(idx, ftab, frag);

  // 6) merger = relu(cond @ Wa + fragment @ Wb + bm)
  wmma_gemm_f32<true, true, false><<<dim3(D_DIM / TN, B_DIM / TM), 256, 0, stream>>>(
      cond, Wa, bm, frag, Wb, nullptr, merger, B_DIM, D_DIM, D_DIM, D_DIM);
}